// TotalLoss_83073257439323
// MI455X (gfx1250) — compile-verified
//
#include <hip/hip_runtime.h>

#define P_   6
#define B_   256
#define C_   8192
#define D_   512
#define PD   3072          // P_*D_
#define ROWS (P_ * B_)     // 1536

typedef __attribute__((ext_vector_type(2))) float v2f;
typedef __attribute__((ext_vector_type(8))) float v8f;

__device__ __forceinline__ float blockReduceSum(float v, float* buf) {
  const int tid = threadIdx.x;
  buf[tid] = v;
  __syncthreads();
  for (int s = 128; s > 0; s >>= 1) {
    if (tid < s) buf[tid] += buf[tid + s];
    __syncthreads();
  }
  float r = buf[0];
  __syncthreads();
  return r;
}

__device__ __forceinline__ float blockReduceMax(float v, float* buf) {
  const int tid = threadIdx.x;
  buf[tid] = v;
  __syncthreads();
  for (int s = 128; s > 0; s >>= 1) {
    if (tid < s) buf[tid] = fmaxf(buf[tid], buf[tid + s]);
    __syncthreads();
  }
  float r = buf[0];
  __syncthreads();
  return r;
}

// -------- Kernel 1: fused label-smoothed CE + KD-KL per (p,b) row over C=8192 --------
// float4 (b128) loads; logits row kept in registers so it is read from HBM exactly once.
__global__ void __launch_bounds__(256) ce_kd_kernel(
    const float* __restrict__ logits,
    const float* __restrict__ soft,
    const float* __restrict__ ew,
    const int* __restrict__ labels,
    float* __restrict__ ce_rows,
    float* __restrict__ kd_rows) {
  __shared__ float buf[256];
  const int row = blockIdx.x;        // p*B + b
  const int p   = row >> 8;
  const int b   = row & 255;
  const float4* __restrict__ x4  = (const float4*)(logits + (size_t)row * C_);
  const float4* __restrict__ sp4 = (const float4*)(soft   + (size_t)row * C_);
  const int tid = threadIdx.x;

  // pass 1: load logits into registers (8 x float4 = 32 elems/thread), row max
  float4 xv[8];
  float m = -3.4e38f;
#pragma unroll
  for (int i = 0; i < 8; ++i) {
    xv[i] = x4[tid + i * 256];
    m = fmaxf(m, fmaxf(fmaxf(xv[i].x, xv[i].y), fmaxf(xv[i].z, xv[i].w)));
  }
  m = blockReduceMax(m, buf);

  // pass 2: stream soft_labels, reuse register-resident logits
  const float invT = 1.0f / 3.0f;    // 1/T_KD
  float s1 = 0.f, s2 = 0.f, sx = 0.f, spx = 0.f, spp = 0.f, sps = 0.f;
#pragma unroll
  for (int i = 0; i < 8; ++i) {
    const float4 pv = sp4[tid + i * 256];
    const float xs[4] = {xv[i].x, xv[i].y, xv[i].z, xv[i].w};
    const float ps[4] = {pv.x, pv.y, pv.z, pv.w};
#pragma unroll
    for (int c = 0; c < 4; ++c) {
      const float xvv = xs[c];
      const float pvv = ps[c];
      s1  += expf(xvv - m);             // for log_softmax(x)
      s2  += expf((xvv - m) * invT);    // for log_softmax(x/T)
      sx  += xvv;
      spx += pvv * xvv;
      spp += (pvv > 0.f) ? pvv * logf(pvv) : 0.f;
      sps += pvv;
    }
  }
  s1  = blockReduceSum(s1, buf);
  s2  = blockReduceSum(s2, buf);
  sx  = blockReduceSum(sx, buf);
  spx = blockReduceSum(spx, buf);
  spp = blockReduceSum(spp, buf);
  sps = blockReduceSum(sps, buf);

  if (tid == 0) {
    const float logZ1 = m + logf(s1);
    const float xl    = logits[(size_t)row * C_ + labels[b]];
    const float nll    = logZ1 - xl;                       // -logp[label]
    const float smooth = logZ1 - sx * (1.0f / (float)C_);  // -mean(logp)
    ce_rows[row] = 0.9f * nll + 0.1f * smooth;

    const float logZ2 = m * invT + logf(s2);
    // sum_c p*logq = spx/T - logZ2*sum(p)
    const float kl = spp - (spx * invT - logZ2 * sps);
    kd_rows[row] = kl * ew[p];
  }
}

// -------- Kernel 2: adversarial CE over [P,B,2] (eps=0) --------
__global__ void adv_kernel(const float* __restrict__ adv,
                           const int* __restrict__ mlab,
                           float* __restrict__ out_sum) {
  __shared__ float buf[256];
  const int tid = threadIdx.x;
  float acc = 0.f;
  for (int r = tid; r < ROWS; r += 256) {
    const int b = r & 255;
    const float2 a = ((const float2*)adv)[r];
    const float m  = fmaxf(a.x, a.y);
    const float logZ = m + logf(expf(a.x - m) + expf(a.y - m));
    const float xl = (mlab[b] != 0) ? a.y : a.x;
    acc += logZ - xl;
  }
  acc = blockReduceSum(acc, buf);
  if (tid == 0) out_sum[0] = acc;
}

// -------- Kernel 3: gather [P,B,D]->[B,PD], L2-normalize rows (b128 loads/stores) --------
__global__ void norm_kernel(const float* __restrict__ feat,
                            float* __restrict__ fn) {
  __shared__ float buf[256];
  const int b = blockIdx.x, tid = threadIdx.x;
  float4 v[3];
  float ss = 0.f;
#pragma unroll
  for (int i = 0; i < 3; ++i) {
    const int idx4 = tid + i * 256;         // 0..767 (float4 index in [PD/4])
    const int p  = idx4 >> 7;               // /128  (512 floats = 128 float4 per part)
    const int d4 = idx4 & 127;
    const float4 t = ((const float4*)feat)[(size_t)(p * B_ + b) * (D_ / 4) + d4];
    v[i] = t;
    ss += t.x * t.x + t.y * t.y + t.z * t.z + t.w * t.w;
  }
  ss = blockReduceSum(ss, buf);
  const float rn = 1.0f / sqrtf(ss);
#pragma unroll
  for (int i = 0; i < 3; ++i) {
    float4 o;
    o.x = v[i].x * rn; o.y = v[i].y * rn; o.z = v[i].z * rn; o.w = v[i].w * rn;
    ((float4*)fn)[(size_t)b * (PD / 4) + tid + i * 256] = o;
  }
}

// -------- Kernel 4: Gram = fn @ fn^T via V_WMMA_F32_16X16X4_F32, one wave per 16x16 tile --------
__global__ void __launch_bounds__(32) gram_kernel(const float* __restrict__ fn,
                                                  float* __restrict__ gram) {
  const int lane = threadIdx.x;
  const int grp  = lane >> 4;      // 0: K=k0,k0+1 ; 1: K=k0+2,k0+3
  const int mr   = lane & 15;      // row within fragment
  const float* __restrict__ arow = fn + (size_t)(blockIdx.y * 16 + mr) * PD + 2 * grp;
  const float* __restrict__ brow = fn + (size_t)(blockIdx.x * 16 + mr) * PD + 2 * grp;

  v8f acc = {0.f, 0.f, 0.f, 0.f, 0.f, 0.f, 0.f, 0.f};
#pragma unroll 4
  for (int k0 = 0; k0 < PD; k0 += 4) {
    v2f a = *(const v2f*)(arow + k0);
    v2f bb = *(const v2f*)(brow + k0);
    // 8 args: (neg_a, A, neg_b, B, c_mod, C, reuse_a, reuse_b)
    acc = __builtin_amdgcn_wmma_f32_16x16x4_f32(false, a, false, bb,
                                                (short)0, acc, false, false);
  }
  const int col = blockIdx.x * 16 + mr;
#pragma unroll
  for (int r = 0; r < 8; ++r) {
    const int rowm = blockIdx.y * 16 + r + 8 * grp;   // C/D layout: VGPR r -> M=r (lanes 0-15), M=r+8 (16-31)
    gram[rowm * 256 + col] = acc[r];
  }
}

// -------- Kernel 5: batch-hard triplet mining from Gram --------
__global__ void triplet_kernel(const float* __restrict__ gram,
                               const int* __restrict__ labels,
                               float* __restrict__ out_sum) {
  __shared__ float buf[256];
  __shared__ int   lab[256];
  __shared__ float ssd[256];
  const int i = threadIdx.x;
  lab[i] = labels[i];
  ssd[i] = gram[i * 257];       // self inner product (== ||f_i||^2 post-norm)
  __syncthreads();

  const int   li  = lab[i];
  const float ssi = ssd[i];
  float maxp = -3.4e38f, minn = 3.4e38f;
  int hasp = 0, hasn = 0;
  for (int j = 0; j < 256; ++j) {
    const float g  = gram[i * 256 + j];
    float d2 = ssi + ssd[j] - 2.0f * g;
    d2 = fmaxf(d2, 0.0f);
    const float dist = (d2 > 0.0f) ? sqrtf(d2) : 0.0f;
    const bool same = (lab[j] == li);
    if (same && (j != i)) { hasp = 1; maxp = fmaxf(maxp, dist); }
    if (!same)            { hasn = 1; minn = fminf(minn, dist); }
  }
  const float ap = hasp ? maxp : 0.0f;
  const float an = hasn ? minn : 1.0e6f;
  float l = fmaxf(ap - an + 0.3f, 0.0f);
  l = blockReduceSum(l, buf);
  if (i == 0) out_sum[0] = l;
}

// -------- Kernel 6: final combine --------
__global__ void final_kernel(const float* __restrict__ ce_rows,
                             const float* __restrict__ kd_rows,
                             const float* __restrict__ adv_sum,
                             const float* __restrict__ tri_sum,
                             const int* __restrict__ epoch,
                             float* __restrict__ out) {
  __shared__ float buf[256];
  const int tid = threadIdx.x;
  float ce = 0.f, kd = 0.f;
  for (int r = tid; r < ROWS; r += 256) { ce += ce_rows[r]; kd += kd_rows[r]; }
  ce = blockReduceSum(ce, buf);
  kd = blockReduceSum(kd, buf);
  if (tid == 0) {
    const float loss_id    = ce / (float)ROWS;
    const float loss_graph = kd * 9.0f / (float)ROWS;   // * T^2 / (P*B)
    const float loss_adv   = adv_sum[0] / (float)ROWS;
    const float loss_tri   = tri_sum[0] / (float)B_;
    const float gate = (epoch[0] >= 20) ? 1.0f : 0.0f;
    out[0] = loss_id + 1.0f * loss_tri + 0.1f * gate * loss_graph + 0.1f * loss_adv;
  }
}

extern "C" void kernel_launch(void* const* d_in, const int* in_sizes, int n_in,
                              void* d_out, int out_size, void* d_ws, size_t ws_size,
                              hipStream_t stream) {
  (void)in_sizes; (void)n_in; (void)out_size; (void)ws_size;
  const float* id_logits   = (const float*)d_in[0];   // [P,B,C]
  const float* id_features = (const float*)d_in[1];   // [P,B,D]
  const float* soft_labels = (const float*)d_in[2];   // [P,B,C]
  const float* ew          = (const float*)d_in[3];   // [P]
  const float* adv_logits  = (const float*)d_in[4];   // [P,B,2]
  const int*   labels      = (const int*)d_in[5];     // [B]
  const int*   mlab        = (const int*)d_in[6];     // [B]
  const int*   epoch       = (const int*)d_in[7];     // [1]

  float* ws      = (float*)d_ws;
  float* fn      = ws;                       // 786432 floats (3 MB)
  float* gram    = fn + (size_t)B_ * PD;     // 65536 floats
  float* ce_rows = gram + 256 * 256;         // 1536
  float* kd_rows = ce_rows + ROWS;           // 1536
  float* adv_sum = kd_rows + ROWS;           // 1
  float* tri_sum = adv_sum + 1;              // 1
  float* out     = (float*)d_out;

  ce_kd_kernel<<<ROWS, 256, 0, stream>>>(id_logits, soft_labels, ew, labels, ce_rows, kd_rows);
  adv_kernel<<<1, 256, 0, stream>>>(adv_logits, mlab, adv_sum);
  norm_kernel<<<B_, 256, 0, stream>>>(id_features, fn);
  gram_kernel<<<dim3(16, 16), 32, 0, stream>>>(fn, gram);
  triplet_kernel<<<1, 256, 0, stream>>>(gram, labels, tri_sum);
  final_kernel<<<1, 256, 0, stream>>>(ce_rows, kd_rows, adv_sum, tri_sum, epoch, out);
}